// GATScatter_56642028700315
// MI455X (gfx1250) — compile-verified
//
#include <hip/hip_runtime.h>
#include <math.h>

typedef __attribute__((ext_vector_type(2))) float v2f;
typedef __attribute__((ext_vector_type(8))) float v8f;

#define DIN 128
#define DOUT 128
#define HH 4
#define DS 32

// ---------------------------------------------------------------------------
// Kernel 1: init accumulators (out=0, denom=0, mx=-inf)
// ---------------------------------------------------------------------------
__global__ void gat_init_kernel(float* __restrict__ out, float* __restrict__ mx,
                                float* __restrict__ denom, int n) {
  size_t stride = (size_t)gridDim.x * blockDim.x;
  size_t i0 = (size_t)blockIdx.x * blockDim.x + threadIdx.x;
  size_t total = (size_t)n * DOUT;
  for (size_t t = i0; t < total; t += stride) out[t] = 0.0f;
  size_t total4 = (size_t)n * HH;
  for (size_t t = i0; t < total4; t += stride) {
    mx[t] = -__builtin_inff();
    denom[t] = 0.0f;
  }
}

// ---------------------------------------------------------------------------
// Kernel 2: feat_src = x@W0^T + b0 ; feat_self = x@W1^T + b1 ; el = lrelu(<feat_src,att>)
// One block = 16 rows of x. 8 waves, each wave owns a 16-col stripe of both outputs.
// WMMA f32 16x16x4, K swept 0..128 in steps of 4. A fragment shared by both GEMMs.
// ---------------------------------------------------------------------------
__global__ __launch_bounds__(256) void gat_gemm_el_kernel(
    const float* __restrict__ x, const float* __restrict__ W0, const float* __restrict__ b0,
    const float* __restrict__ W1, const float* __restrict__ b1, const float* __restrict__ att,
    float* __restrict__ feat_src, float* __restrict__ feat_self, float* __restrict__ el,
    int n) {
  __shared__ float xs[16][132];  // stride 132: 132%64==4 -> conflict-free column reads
  const int tid = threadIdx.x;
  const int lane = tid & 31;
  const int wave = tid >> 5;
  const int row_base = blockIdx.x * 16;

  // cooperatively stage the 16x128 x-tile into LDS (float4 global loads)
#pragma unroll
  for (int i = 0; i < 2; ++i) {
    int f4 = tid + i * 256;       // 0..511 float4 chunks
    int r = f4 >> 5;              // row 0..15
    int c = (f4 & 31) * 4;        // col 0..124
    int rr = row_base + r;
    if (rr >= n) rr = n - 1;      // clamp (N is a multiple of 16 in practice)
    float4 v = *(const float4*)(x + (size_t)rr * DIN + c);
    xs[r][c + 0] = v.x; xs[r][c + 1] = v.y; xs[r][c + 2] = v.z; xs[r][c + 3] = v.w;
  }
  __syncthreads();

  const int m = lane & 15;        // A row / B-C-D column-within-tile
  const int g = lane >> 4;        // lane half
  const int n_col = wave * 16 + m;

  v8f acc0 = {};
  v8f acc1 = {};
#pragma unroll
  for (int kb = 0; kb < DIN; kb += 4) {
    const int k0 = kb + 2 * g;
    v2f a;  a.x  = xs[m][k0];                    a.y  = xs[m][k0 + 1];
    v2f bb0; bb0.x = W0[(size_t)n_col * DIN + k0]; bb0.y = W0[(size_t)n_col * DIN + k0 + 1];
    v2f bb1; bb1.x = W1[(size_t)n_col * DIN + k0]; bb1.y = W1[(size_t)n_col * DIN + k0 + 1];
    acc0 = __builtin_amdgcn_wmma_f32_16x16x4_f32(false, a, false, bb0, (short)0, acc0, false, false);
    acc1 = __builtin_amdgcn_wmma_f32_16x16x4_f32(false, a, false, bb1, (short)0, acc1, false, false);
  }

  const float bias0 = b0[n_col];
  const float bias1 = b1[n_col];
#pragma unroll
  for (int i = 0; i < 8; ++i) {
    int r = i + 8 * g;            // C/D: VGPR i holds rows i (lanes 0-15) and i+8 (lanes 16-31)
    int rr = row_base + r;
    if (rr < n) {
      size_t o = (size_t)rr * DOUT + n_col;
      feat_src[o]  = acc0[i] + bias0;
      feat_self[o] = acc1[i] + bias1;
    }
  }

  // reuse LDS: stash feat_src tile, then per-(row,head) attention dot + LeakyReLU(0.2)
  __syncthreads();
#pragma unroll
  for (int i = 0; i < 8; ++i) {
    int r = i + 8 * g;
    xs[r][n_col] = acc0[i] + bias0;
  }
  __syncthreads();
  if (tid < 64) {
    int r = tid >> 2;
    int h = tid & 3;
    float s = 0.0f;
#pragma unroll
    for (int d = 0; d < DS; ++d) s += xs[r][h * DS + d] * att[h * DS + d];
    float e = (s > 0.0f) ? s : 0.2f * s;
    int rr = row_base + r;
    if (rr < n) el[(size_t)rr * HH + h] = e;
  }
}

// ---------------------------------------------------------------------------
// Kernel 3: segment max: mx[rows[e], h] = max over edges of el[cols[e], h]
// float atomic-max via signed-int max / unsigned-int min trick (init = -inf).
// ---------------------------------------------------------------------------
__device__ __forceinline__ void atomicMaxF(float* addr, float v) {
  if (v >= 0.0f)
    atomicMax((int*)addr, __float_as_int(v));
  else
    atomicMin((unsigned int*)addr, __float_as_uint(v));
}

__global__ __launch_bounds__(256) void gat_edge_max_kernel(
    const int* __restrict__ rows, const int* __restrict__ cols,
    const float* __restrict__ el, float* __restrict__ mx, int e4) {
  int i = blockIdx.x * blockDim.x + threadIdx.x;
  if (i >= e4) return;
  int e = i >> 2;
  int h = i & 3;
  atomicMaxF(&mx[(size_t)rows[e] * HH + h], el[(size_t)cols[e] * HH + h]);
}

// ---------------------------------------------------------------------------
// Kernel 4: one wave per edge. ev[h]=exp(el[c,h]-mx[r,h])*val; denom += ev;
// out[r, h*32+d] += ev[h]*feat_src[c, h*32+d]  (128 coalesced f32 atomics/edge;
// num + feat_src both live in the 192MB L2, so this runs at L2 rate).
// ---------------------------------------------------------------------------
__global__ __launch_bounds__(256) void gat_aggregate_kernel(
    const int* __restrict__ rows, const int* __restrict__ cols,
    const float* __restrict__ vals, const float* __restrict__ el,
    const float* __restrict__ mx, const float* __restrict__ feat_src,
    float* __restrict__ denom, float* __restrict__ out, int e_cnt) {
  int w = (int)(((size_t)blockIdx.x * blockDim.x + threadIdx.x) >> 5);
  int lane = threadIdx.x & 31;
  if (w >= e_cnt) return;
  const int r = rows[w];
  const int c = cols[w];
  const float val = vals[w];
  const float4 e4 = *(const float4*)(el + (size_t)c * HH);
  const float4 m4 = *(const float4*)(mx + (size_t)r * HH);
  const float ev0 = expf(e4.x - m4.x) * val;
  const float ev1 = expf(e4.y - m4.y) * val;
  const float ev2 = expf(e4.z - m4.z) * val;
  const float ev3 = expf(e4.w - m4.w) * val;
  if (lane == 0) {
    atomicAdd(&denom[(size_t)r * HH + 0], ev0);
    atomicAdd(&denom[(size_t)r * HH + 1], ev1);
    atomicAdd(&denom[(size_t)r * HH + 2], ev2);
    atomicAdd(&denom[(size_t)r * HH + 3], ev3);
  }
  const float* __restrict__ fs = feat_src + (size_t)c * DOUT;
  float* __restrict__ o = out + (size_t)r * DOUT;
  atomicAdd(&o[lane +  0], ev0 * fs[lane +  0]);
  atomicAdd(&o[lane + 32], ev1 * fs[lane + 32]);
  atomicAdd(&o[lane + 64], ev2 * fs[lane + 64]);
  atomicAdd(&o[lane + 96], ev3 * fs[lane + 96]);
}

// ---------------------------------------------------------------------------
// Kernel 5: one wave per node. aggr = num/denom; h = relu(aggr + feat_self);
// per-node mean/var via wave32 shfl_xor reduction; scale/offset; write out.
// ---------------------------------------------------------------------------
__global__ __launch_bounds__(256) void gat_finalize_kernel(
    const float* __restrict__ feat_self, const float* __restrict__ denom,
    const float* __restrict__ scale, const float* __restrict__ offset,
    float* __restrict__ out, int n) {
  int node = (int)(((size_t)blockIdx.x * blockDim.x + threadIdx.x) >> 5);
  int lane = threadIdx.x & 31;
  if (node >= n) return;
  float* __restrict__ o = out + (size_t)node * DOUT;
  const float* __restrict__ fsl = feat_self + (size_t)node * DOUT;
  const float4 d4 = *(const float4*)(denom + (size_t)node * HH);
  float dn[4];
  dn[0] = fmaxf(d4.x, 1e-10f);
  dn[1] = fmaxf(d4.y, 1e-10f);
  dn[2] = fmaxf(d4.z, 1e-10f);
  dn[3] = fmaxf(d4.w, 1e-10f);
  float hv[4];
  float s = 0.0f, s2 = 0.0f;
#pragma unroll
  for (int i = 0; i < 4; ++i) {
    int d = i * 32 + lane;        // head = i
    float aggr = o[d] / dn[i];
    float v = aggr + fsl[d];
    v = fmaxf(v, 0.0f);
    hv[i] = v;
    s += v;
    s2 += v * v;
  }
#pragma unroll
  for (int off = 16; off >= 1; off >>= 1) {
    s  += __shfl_xor(s, off, 32);
    s2 += __shfl_xor(s2, off, 32);
  }
  const float mean = s * (1.0f / 128.0f);
  const float var = s2 * (1.0f / 128.0f) - mean * mean;
  const float inv = rsqrtf(var + 1e-9f);
#pragma unroll
  for (int i = 0; i < 4; ++i) {
    int d = i * 32 + lane;
    o[d] = (hv[i] - mean) * inv * scale[d] + offset[d];
  }
}

// ---------------------------------------------------------------------------
extern "C" void kernel_launch(void* const* d_in, const int* in_sizes, int n_in,
                              void* d_out, int out_size, void* d_ws, size_t ws_size,
                              hipStream_t stream) {
  const float* x     = (const float*)d_in[0];
  const int*   rows  = (const int*)d_in[1];
  const int*   cols  = (const int*)d_in[2];
  const float* vals  = (const float*)d_in[3];
  const float* W0    = (const float*)d_in[4];
  const float* b0    = (const float*)d_in[5];
  const float* W1    = (const float*)d_in[6];
  const float* b1    = (const float*)d_in[7];
  const float* att   = (const float*)d_in[8];
  const float* scale = (const float*)d_in[9];
  const float* offs  = (const float*)d_in[10];

  const int n = in_sizes[0] / DIN;   // 100000
  const int e = in_sizes[1];         // 1600000

  // workspace carve (all offsets 16B-aligned): ~107 MB total
  float* ws        = (float*)d_ws;
  float* feat_src  = ws;                               // n*128
  float* feat_self = feat_src + (size_t)n * DOUT;      // n*128
  float* el        = feat_self + (size_t)n * DOUT;     // n*4
  float* mx        = el + (size_t)n * HH;              // n*4
  float* denom     = mx + (size_t)n * HH;              // n*4
  float* out       = (float*)d_out;                    // n*128 (num accumulator, then final)

  gat_init_kernel<<<dim3(2048), dim3(256), 0, stream>>>(out, mx, denom, n);

  gat_gemm_el_kernel<<<dim3((n + 15) / 16), dim3(256), 0, stream>>>(
      x, W0, b0, W1, b1, att, feat_src, feat_self, el, n);

  gat_edge_max_kernel<<<dim3((e * 4 + 255) / 256), dim3(256), 0, stream>>>(
      rows, cols, el, mx, e * 4);

  gat_aggregate_kernel<<<dim3((e + 7) / 8), dim3(256), 0, stream>>>(
      rows, cols, vals, el, mx, feat_src, denom, out, e);

  gat_finalize_kernel<<<dim3((n + 7) / 8), dim3(256), 0, stream>>>(
      feat_self, denom, scale, offs, out, n);
}